// MegaMultiDimensionDampedEma_55327768708061
// MI455X (gfx1250) — compile-verified
//
#include <hip/hip_runtime.h>

// Problem constants
#define L_SEQ   2048
#define BATCH   8
#define DHID    1024
#define NDIM    16
#define SEG     64          // scan segment (4 WMMA row-blocks)
#define NSEG    32          // 2048/64
#define NSPAIR  16          // segment pairs (2 segs = 16 WMMA columns)

// Workspace layout (float offsets). 23 per-d 16x16 matrices, each [1024][256]:
//  0      : ACOMB   combined intra-block Toeplitz (K1 lower + K2 upper)
//  1..3   : AF_dt   fwd Toeplitz, lag 16*dt      AF[i][j] = K1[16dt+i-j]
//  4..6   : AB_dt   bwd Toeplitz, lag 16*dt      AB[i][j] = K2[16dt+j-i]
//  7..10  : E1_ti   fwd carry inject             E1[i][n] = w1_n q1_n^{16ti+i+1}
//  11..14 : E2_ti   bwd carry inject             E2[i][n] = w2_n q2_n^{64-16ti-i}
//  15..18 : F1_tj   fwd carry gen                F1[n][j] = q1_n^{63-16tj-j}
//  19..22 : F2_tj   bwd carry gen                F2[n][j] = q2_n^{16tj+j}
#define MS        262144u
#define WS_QTF    6029312u   // [1024][16] q1^64
#define WS_QTB    6045696u   // [1024][16] q2^64
#define WS_CSF    6062080u   // [32 g][1024 d][16 n][8 b] carries -> (in place) fwd states
#define WS_CSB    10256384u  // backward
#define SEG_STRIDE 131072u   // DHID*NDIM*BATCH
// total ws = 14,450,688 floats (~58 MB)

typedef __attribute__((ext_vector_type(2))) float v2f;  // 16x4 f32 A/B fragment (2 VGPRs)
typedef __attribute__((ext_vector_type(8))) float v8f;  // 16x16 f32 C/D fragment (8 VGPRs)

__device__ __forceinline__ float sigmoidf_(float x) { return 1.0f / (1.0f + __expf(-x)); }

// x tile [128 t][8 b][32 d]: stride-33 pad + XOR swizzle on t-bits whose strides
// alias LDS banks mod 64 (t diff 8 -> 33*64 = 0 mod 64; t diff 64 likewise).
__device__ __forceinline__ int xIdx(int t, int b, int dl) {
    int sw = (((t >> 3) & 1) << 3) ^ (((t >> 6) & 1) << 4);
    return (t * 8 + b) * 33 + (dl ^ sw);
}
// conv-output transpose buffer [32 row][8 b][32 d], same aliasing pattern on row 8/16
__device__ __forceinline__ int cbIdx(int row, int b, int dl) {
    int sw = (((row >> 3) & 1) << 3) ^ (((row >> 4) & 1) << 4);
    return (row * 8 + b) * 33 + (dl ^ sw);
}

// ---------------------------------------------------------------------------
// Kernel 1: per-d operand matrices + q^64 via LDS power tables.
// grid 1024 (block per d), block 256 (thread per matrix element)
// ---------------------------------------------------------------------------
__global__ void ema_precompute(const float* __restrict__ damp,
                               const float* __restrict__ decay,
                               const float* __restrict__ ema,
                               const float* __restrict__ proj,
                               float* __restrict__ ws) {
    const int d = blockIdx.x;
    const int t = threadIdx.x;
    const int i = t >> 4, j = t & 15;
    __shared__ float q1p[16][66], q2p[16][66], w1s[16], w2s[16];
    if (t < 16) {
        int n = t;
        float p1 = sigmoidf_(damp[d * NDIM + n]);
        float q1 = 1.0f - p1 * sigmoidf_(decay[d * NDIM + n]);
        w1s[n] = p1 * ema[d * NDIM + n] * proj[d * NDIM + n] * 0.25f;  // scale=1/sqrt(16)
        int d2 = d + DHID;
        float p2 = sigmoidf_(damp[d2 * NDIM + n]);
        float q2 = 1.0f - p2 * sigmoidf_(decay[d2 * NDIM + n]);
        w2s[n] = p2 * ema[d2 * NDIM + n] * proj[d2 * NDIM + n] * 0.25f;
        float a1 = 1.0f, a2 = 1.0f;
        for (int m = 0; m <= 64; ++m) { q1p[n][m] = a1; q2p[n][m] = a2; a1 *= q1; a2 *= q2; }
    }
    __syncthreads();
    float* wd = ws + d * 256 + t;
    // ACOMB (diagonal legitimately gets K1[0]+K2[0], matching the reference overlap)
    {
        float a = 0.0f;
        if (i >= j) { int m = i - j; for (int n = 0; n < 16; ++n) a += w1s[n] * q1p[n][m]; }
        if (i <= j) { int m = j - i; for (int n = 0; n < 16; ++n) a += w2s[n] * q2p[n][m]; }
        wd[0] = a;
    }
#pragma unroll
    for (int dt = 1; dt <= 3; ++dt) {   // AF_dt, AB_dt
        float a = 0.0f, b = 0.0f;
        for (int n = 0; n < 16; ++n) {
            a += w1s[n] * q1p[n][16 * dt + i - j];
            b += w2s[n] * q2p[n][16 * dt + j - i];
        }
        wd[(size_t)dt * MS] = a;
        wd[(size_t)(3 + dt) * MS] = b;
    }
#pragma unroll
    for (int t4 = 0; t4 < 4; ++t4) {
        wd[(size_t)(7 + t4) * MS]  = w1s[j] * q1p[j][16 * t4 + i + 1];   // E1_t
        wd[(size_t)(11 + t4) * MS] = w2s[j] * q2p[j][64 - 16 * t4 - i];  // E2_t
        wd[(size_t)(15 + t4) * MS] = q1p[i][63 - 16 * t4 - j];           // F1_t
        wd[(size_t)(19 + t4) * MS] = q2p[i][16 * t4 + j];                // F2_t
    }
    if (t < 16) {
        ws[WS_QTF + d * 16 + t] = q1p[t][64];
        ws[WS_QTB + d * 16 + t] = q2p[t][64];
    }
}

// ---------------------------------------------------------------------------
// Kernel 2: segment carries  C_f = SUM_tj F1_tj@X_tj,  C_b = SUM_tj F2_tj@X_tj
// grid (32 d-blocks, 8 pair-groups), block 1024 = 32 waves; wave w owns d = d0+w.
// Carries need only masked x, so the mask is applied once at staging time.
// F fragments are loaded per use from L2 (matrix set is L2-resident, no spills).
// ---------------------------------------------------------------------------
__global__ void __launch_bounds__(1024)
ema_seg_carries(const float* __restrict__ x, const int* __restrict__ mask,
                float* __restrict__ ws) {
    __shared__ float xm[33792];   // masked x tile [128][8][32]
    const int d0 = blockIdx.x * 32;
    const int tid = threadIdx.x;
    const int lane = tid & 31, wave = tid >> 5;
    const int d = d0 + wave;
    const int col = lane & 15, kh = lane >> 4;
    const int half = col >> 3;    // which segment of the pair this column serves
    const int bb = col & 7;
    const int mbase = d * 256 + (lane & 15) * 16 + 2 * kh;

    const int sdl = tid & 31, sb = (tid >> 5) & 7, st0 = tid >> 8;

    for (int pair = blockIdx.y * 2; pair < blockIdx.y * 2 + 2; ++pair) {
#pragma unroll
        for (int k = 0; k < 32; ++k) {   // coalesced stage (consecutive tids -> consecutive d)
            int t = k * 4 + st0;
            int l = pair * 128 + t;
            float v = x[(size_t)l * (BATCH * DHID) + sb * DHID + d0 + sdl];
            xm[xIdx(t, sb, sdl)] = v * (float)mask[sb * L_SEQ + l];
        }
        __syncthreads();

        v8f cf = {}; v8f cbk = {};
#pragma unroll
        for (int tj = 0; tj < 4; ++tj) {
            v2f bx[4];                 // all four B fragments first: independent LDS loads
#pragma unroll
            for (int s = 0; s < 4; ++s) {
                int t0 = half * 64 + tj * 16 + 4 * s + 2 * kh;
                bx[s].x = xm[xIdx(t0,     bb, wave)];
                bx[s].y = xm[xIdx(t0 + 1, bb, wave)];
            }
            const float* p1 = ws + (size_t)(15 + tj) * MS + mbase;
            const float* p2 = ws + (size_t)(19 + tj) * MS + mbase;
#pragma unroll
            for (int s = 0; s < 4; ++s) {
                v2f f1 = *(const v2f*)(p1 + 4 * s);
                v2f f2 = *(const v2f*)(p2 + 4 * s);
                cf  = __builtin_amdgcn_wmma_f32_16x16x4_f32(false, f1, false, bx[s], (short)0, cf,  false, false);
                cbk = __builtin_amdgcn_wmma_f32_16x16x4_f32(false, f2, false, bx[s], (short)0, cbk, false, false);
            }
        }
        const int g = pair * 2 + half;
        const size_t base = (size_t)g * SEG_STRIDE + (size_t)d * (NDIM * BATCH) + bb;
#pragma unroll
        for (int r = 0; r < 8; ++r) {    // D frag: VGPR r holds M=r (lanes 0-15) / r+8 (16-31)
            int n = r + 8 * kh;
            ws[WS_CSF + base + n * 8] = cf[r];
            ws[WS_CSB + base + n * 8] = cbk[r];
        }
        __syncthreads();
    }
}

// ---------------------------------------------------------------------------
// Kernel 3: sequential scan over 32 segments, in place: carries -> incoming states.
// 131072 threads, one per (d,n,b); [g][d][n][b] layout -> fully coalesced.
// ---------------------------------------------------------------------------
__global__ void ema_seg_scan(float* __restrict__ ws) {
    const int tid = blockIdx.x * blockDim.x + threadIdx.x;  // 0..131071
    const int d = tid >> 7, n = (tid >> 3) & 15;
    float* pf = ws + WS_CSF + tid;
    float* pb = ws + WS_CSB + tid;
    const float qtf = ws[WS_QTF + d * 16 + n];
    const float qtb = ws[WS_QTB + d * 16 + n];
    float S = 0.0f;
    for (int g = 0; g < NSEG; ++g) {          // Sin_f[g] = state entering segment g
        float tmp = pf[(size_t)g * SEG_STRIDE]; pf[(size_t)g * SEG_STRIDE] = S; S = qtf * S + tmp;
    }
    S = 0.0f;
    for (int g = NSEG - 1; g >= 0; --g) {     // Sin_b[g] = state entering from the right
        float tmp = pb[(size_t)g * SEG_STRIDE]; pb[(size_t)g * SEG_STRIDE] = S; S = qtb * S + tmp;
    }
}

// ---------------------------------------------------------------------------
// Kernel 4: out = silu( SUM_tj M_{ti,tj}@X_tj + E1_ti@Sf + E2_ti@Sb + rw*x )
// Stages RAW x (mask applied at B-fragment read from a small mask tile) so the
// residual can read unmasked x straight from LDS. Conv fragments funnel through
// a 33KB LDS transpose buffer (4 rounds/pair) for 128B-coalesced global stores.
// ---------------------------------------------------------------------------
__global__ void __launch_bounds__(1024)
ema_output(const float* __restrict__ x, const int* __restrict__ mask,
           const float* __restrict__ rw, const float* __restrict__ ws,
           float* __restrict__ out) {
    __shared__ float xm[33792];   // raw x tile
    __shared__ float cb[8448];    // conv transpose buffer [32 row][8 b][32 d]
    __shared__ float mk[1152];    // mask tile [128][8] (stride 9)
    const int d0 = blockIdx.x * 32;
    const int tid = threadIdx.x;
    const int lane = tid & 31, wave = tid >> 5;
    const int d = d0 + wave;
    const int col = lane & 15, kh = lane >> 4;
    const int half = col >> 3;
    const int bb = col & 7;
    const int mbase = d * 256 + (lane & 15) * 16 + 2 * kh;

    const int sdl = tid & 31, sb = (tid >> 5) & 7, st0 = tid >> 8;
    const float rws = rw[d0 + sdl];

    for (int pair = blockIdx.y * 2; pair < blockIdx.y * 2 + 2; ++pair) {
#pragma unroll
        for (int k = 0; k < 32; ++k) {
            int t = k * 4 + st0;
            xm[xIdx(t, sb, sdl)] = x[(size_t)(pair * 128 + t) * (BATCH * DHID) + sb * DHID + d0 + sdl];
        }
        { int tt = tid >> 3, b = tid & 7;
          mk[tt * 9 + b] = (float)mask[b * L_SEQ + pair * 128 + tt]; }
        __syncthreads();

        v8f acc[4] = {};
#pragma unroll
        for (int tj = 0; tj < 4; ++tj) {
            v2f bx[4];
#pragma unroll
            for (int s = 0; s < 4; ++s) {
                int t0 = half * 64 + tj * 16 + 4 * s + 2 * kh;
                bx[s].x = xm[xIdx(t0,     bb, wave)] * mk[t0 * 9 + bb];
                bx[s].y = xm[xIdx(t0 + 1, bb, wave)] * mk[(t0 + 1) * 9 + bb];
            }
#pragma unroll
            for (int ti = 0; ti < 4; ++ti) {
                const int dt = ti - tj;
                const int mIdx = (dt == 0) ? 0 : (dt > 0 ? dt : 3 - dt);  // ACOMB / AF_dt / AB_dt
                const float* mp = ws + (size_t)mIdx * MS + mbase;
#pragma unroll
                for (int s = 0; s < 4; ++s) {
                    v2f mf = *(const v2f*)(mp + 4 * s);
                    acc[ti] = __builtin_amdgcn_wmma_f32_16x16x4_f32(false, mf, false, bx[s], (short)0, acc[ti], false, false);
                }
            }
        }

        // incoming-state B fragments for this column's segment (K = n); loaded
        // here (not earlier) so they are not live through the matmul loop above
        const int g = pair * 2 + half;
        const size_t cbase = (size_t)g * SEG_STRIDE + (size_t)d * (NDIM * BATCH) + bb;
        v2f bf[4], bw[4];
#pragma unroll
        for (int s = 0; s < 4; ++s) {
            int k0 = 4 * s + 2 * kh;
            bf[s].x = ws[WS_CSF + cbase + (size_t)k0 * 8];
            bf[s].y = ws[WS_CSF + cbase + (size_t)(k0 + 1) * 8];
            bw[s].x = ws[WS_CSB + cbase + (size_t)k0 * 8];
            bw[s].y = ws[WS_CSB + cbase + (size_t)(k0 + 1) * 8];
        }
#pragma unroll
        for (int ti = 0; ti < 4; ++ti) {     // carry injection
            const float* e1p = ws + (size_t)(7 + ti) * MS + mbase;
            const float* e2p = ws + (size_t)(11 + ti) * MS + mbase;
#pragma unroll
            for (int s = 0; s < 4; ++s) {
                v2f e1 = *(const v2f*)(e1p + 4 * s);
                v2f e2 = *(const v2f*)(e2p + 4 * s);
                acc[ti] = __builtin_amdgcn_wmma_f32_16x16x4_f32(false, e1, false, bf[s], (short)0, acc[ti], false, false);
                acc[ti] = __builtin_amdgcn_wmma_f32_16x16x4_f32(false, e2, false, bw[s], (short)0, acc[ti], false, false);
            }
        }

        // 4 transpose+store rounds (xm stays intact: residual reads raw x from it)
#pragma unroll
        for (int ti = 0; ti < 4; ++ti) {
#pragma unroll
            for (int r = 0; r < 8; ++r) {
                int row = half * 16 + r + 8 * kh;
                cb[cbIdx(row, bb, wave)] = acc[ti][r];
            }
            __syncthreads();
#pragma unroll
            for (int k = 0; k < 8; ++k) {
                int idx = st0 * 8 + k;         // 0..31
                int hh = idx >> 4, rr = idx & 15;
                int t = hh * 64 + ti * 16 + rr;
                float z = cb[cbIdx(hh * 16 + rr, sb, sdl)] + xm[xIdx(t, sb, sdl)] * rws;
                out[(size_t)(pair * 128 + t) * (BATCH * DHID) + sb * DHID + d0 + sdl] =
                    z / (1.0f + __expf(-z));   // silu
            }
            __syncthreads();
        }
    }
}

// ---------------------------------------------------------------------------
extern "C" void kernel_launch(void* const* d_in, const int* in_sizes, int n_in,
                              void* d_out, int out_size, void* d_ws, size_t ws_size,
                              hipStream_t stream) {
    const float* x     = (const float*)d_in[0];  // (2048, 8, 1024) f32
    const float* damp  = (const float*)d_in[1];  // (2048, 16, 1)
    const float* decay = (const float*)d_in[2];  // (2048, 16, 1)
    const float* ema   = (const float*)d_in[3];  // (2048, 16, 1)
    const float* proj  = (const float*)d_in[4];  // (2048, 16)
    const float* rw    = (const float*)d_in[5];  // (1024,)
    const int*   mask  = (const int*)d_in[6];    // (8, 2048) int32
    float* ws  = (float*)d_ws;
    float* out = (float*)d_out;

    ema_precompute<<<1024, 256, 0, stream>>>(damp, decay, ema, proj, ws);
    ema_seg_carries<<<dim3(32, 8), 1024, 0, stream>>>(x, mask, ws);
    ema_seg_scan<<<512, 256, 0, stream>>>(ws);
    ema_output<<<dim3(32, 8), 1024, 0, stream>>>(x, mask, rw, ws, out);
}